// loss_guided_gpu_33182917329297
// MI455X (gfx1250) — compile-verified
//
#include <hip/hip_runtime.h>
#include <stdint.h>

// ---------------------------------------------------------------------------
// Self-guided-filter L1 loss, fused single pass.
//   shape (8,3,512,512) fp32, radius 5, eps 1e-6, zero-padded box / 121.
// Bandwidth analysis: 75.5 MB in -> 4 B out (HBM floor ~3.2us @23.3TB/s, and
// the whole working set is L2-resident at 192MB). The stencil math is made
// cheap with sliding-window running sums (2 LDS reads + 2 adds per output
// instead of 11-22 taps), so LDS/VALU no longer dominate over HBM.
// CDNA5 path: async global->LDS B128 copies (ASYNCcnt) stage 52x64 tiles.
// ---------------------------------------------------------------------------

#define RAD      5
#define KDIA     11                 // 2*RAD+1
#define TILE     32
#define HALO     (2 * RAD)          // 10: total halo of the two-stage stencil
#define REG      (TILE + 4 * RAD)   // 52: input region rows
#define XW       64                 // staged region width (aligned float4s)
#define XOFF     16                 // global col of xs[.][0] is px0 - XOFF
#define XSTR     72                 // xs row stride: 16B-aligned, 72%64=8
#define MID      (TILE + 2 * RAD)   // 42: a/b region side
#define MIDS     (MID + 1)          // 43
#define TS       (TILE + 1)         // 33
#define IMG_H    512
#define IMG_W    512
#define NPLANES  24                 // 8 * 3
#define TILES_X  (IMG_W / TILE)     // 16
#define TILES_Y  (IMG_H / TILE)     // 16
#define NTHREADS 256
#define INV_AREA (1.0f / 121.0f)
#define EPSF     1e-6f
#define INV_N    (1.0f / (float)(NPLANES * IMG_H * IMG_W))

#if defined(__has_builtin)
#if __has_builtin(__builtin_amdgcn_global_load_async_to_lds_b128) && \
    __has_builtin(__builtin_amdgcn_s_wait_asynccnt)
#define USE_ASYNC_B128 1
#elif __has_builtin(__builtin_amdgcn_global_load_async_to_lds_b32) && \
    __has_builtin(__builtin_amdgcn_s_wait_asynccnt)
#define USE_ASYNC_B32 1
#endif
#endif

typedef int v4i __attribute__((ext_vector_type(4)));
typedef __attribute__((address_space(1))) int global_i32;
typedef __attribute__((address_space(3))) int lds_i32;
typedef __attribute__((address_space(1))) v4i global_v4i;
typedef __attribute__((address_space(3))) v4i lds_v4i;

// One self-guided-filter pass over a 32x32 tile of one (512,512) plane.
// Thread t emits 4 outputs: col = t/8, rows = (t%8)*4 .. +3 (same mapping
// every call, so the caller can combine results in registers).
__device__ __forceinline__ void filter_tile(
    const float* __restrict__ plane, int py0, int px0, int tid, bool interior,
    float (*xs)[XSTR], float (*hx)[MIDS], float (*hxx)[MIDS],
    float (*aa)[MIDS], float (*bb)[MIDS],
    float (*hA)[TS], float (*hB)[TS],
    float fout[4])
{
    __syncthreads();  // WAR: previous pass may still be reading the buffers

    // ---- stage 0: global -> LDS, 52 rows x 64 cols, zero outside image ----
    // local (iy, ix) <-> global (py0 - HALO + iy, px0 - XOFF + ix).
    // float4 slots: 512 and the region width are multiples of 4, so a float4
    // is either fully inside or fully outside the image (never straddles).
#if defined(USE_ASYNC_B128)
    const int NV4 = REG * (XW / 4);  // 832 float4 slots
    for (int base = 0; base < NV4; base += NTHREADS) {  // uniform trip, full EXEC
        int idx = base + tid;
        if (idx > NV4 - 1) idx = NV4 - 1;               // benign duplicate DMA
        int iy = idx >> 4, ix = (idx & 15) << 2;
        int gy = py0 - HALO + iy, gx = px0 - XOFF + ix;
        if (!interior) {
            gy = min(max(gy, 0), IMG_H - 1);
            if (gx < 0 || gx >= IMG_W) gx = 0;          // fully-OOB slot: safe addr
        }
        __builtin_amdgcn_global_load_async_to_lds_b128(
            (global_v4i*)(plane + (size_t)gy * IMG_W + gx),
            (lds_v4i*)&xs[iy][ix], 0, 0);
    }
    __builtin_amdgcn_s_wait_asynccnt(0);
    __syncthreads();
#elif defined(USE_ASYNC_B32)
    const int NLD = REG * XW;  // 3328
    for (int base = 0; base < NLD; base += NTHREADS) {
        int idx = base + tid;
        if (idx > NLD - 1) idx = NLD - 1;
        int iy = idx >> 6, ix = idx & 63;
        int gy = py0 - HALO + iy, gx = px0 - XOFF + ix;
        if (!interior) {
            gy = min(max(gy, 0), IMG_H - 1);
            gx = min(max(gx, 0), IMG_W - 1);
        }
        __builtin_amdgcn_global_load_async_to_lds_b32(
            (global_i32*)(plane + (size_t)gy * IMG_W + gx),
            (lds_i32*)&xs[iy][ix], 0, 0);
    }
    __builtin_amdgcn_s_wait_asynccnt(0);
    __syncthreads();
#else
    for (int idx = tid; idx < REG * XW; idx += NTHREADS) {
        int iy = idx >> 6, ix = idx & 63;
        int gy = py0 - HALO + iy, gx = px0 - XOFF + ix;
        bool inb = (gy >= 0) && (gy < IMG_H) && (gx >= 0) && (gx < IMG_W);
        xs[iy][ix] = inb ? plane[(size_t)gy * IMG_W + gx] : 0.0f;
    }
#endif
#if defined(USE_ASYNC_B128) || defined(USE_ASYNC_B32)
    if (!interior) {  // zero the cells whose true coordinates are off-image
        for (int idx = tid; idx < REG * XW; idx += NTHREADS) {
            int iy = idx >> 6, ix = idx & 63;
            int gy = py0 - HALO + iy, gx = px0 - XOFF + ix;
            if (gy < 0 || gy >= IMG_H || gx < 0 || gx >= IMG_W)
                xs[iy][ix] = 0.0f;
        }
    }
#endif
    __syncthreads();

    // ---- stage 1 horizontal: sliding 11-sums of x, x^2 ----
    // 52 rows x 42 out cols; 3 segments of 14 per row (156 threads).
    // out col j covers xs cols j+6 .. j+16 (center = global px0 + j - 5).
    if (tid < REG * 3) {
        int row = tid / 3, seg = tid - row * 3;
        int j0 = seg * 14;
        float s = 0.0f, s2 = 0.0f;
#pragma unroll
        for (int d = 0; d < KDIA; ++d) {
            float v = xs[row][j0 + 6 + d];
            s += v; s2 = fmaf(v, v, s2);
        }
        hx[row][j0] = s; hxx[row][j0] = s2;
#pragma unroll
        for (int j = 1; j < 14; ++j) {
            float vn = xs[row][j0 + j + 16];   // entering tap
            float vo = xs[row][j0 + j + 5];    // leaving tap
            s += vn - vo;
            s2 += fmaf(vn, vn, -(vo * vo));
            hx[row][j0 + j] = s; hxx[row][j0 + j] = s2;
        }
    }
    __syncthreads();

    // ---- stage 1 vertical: sliding sums -> mean/corr -> a, b (42x42) ----
    // 42 cols x 6 segments of 7 rows (252 threads). out row r uses hx r..r+10.
    if (tid < MID * 6) {
        int col = tid / 6, seg = tid - col * 6;
        int r0 = seg * 7;
        float s = 0.0f, s2 = 0.0f;
#pragma unroll
        for (int d = 0; d < KDIA; ++d) { s += hx[r0 + d][col]; s2 += hxx[r0 + d][col]; }
#pragma unroll
        for (int r = 0; r < 7; ++r) {
            if (r) {
                s  += hx[r0 + r + 10][col]  - hx[r0 + r - 1][col];
                s2 += hxx[r0 + r + 10][col] - hxx[r0 + r - 1][col];
            }
            float mean = s * INV_AREA;
            float corr = s2 * INV_AREA;
            float var = corr - mean * mean;
            float A = var / (var + EPSF);
            aa[r0 + r][col] = A;
            bb[r0 + r][col] = mean - A * mean;
        }
    }
    __syncthreads();

    // ---- stage 2 horizontal: sliding sums of a, b (42 rows x 32 cols) ----
    // 4 segments of 8 per row (168 threads). out col m uses a cols m..m+10.
    if (tid < MID * 4) {
        int row = tid >> 2, seg = tid & 3;
        int m0 = seg * 8;
        float sA = 0.0f, sB = 0.0f;
#pragma unroll
        for (int d = 0; d < KDIA; ++d) { sA += aa[row][m0 + d]; sB += bb[row][m0 + d]; }
        hA[row][m0] = sA; hB[row][m0] = sB;
#pragma unroll
        for (int m = 1; m < 8; ++m) {
            sA += aa[row][m0 + m + 10] - aa[row][m0 + m - 1];
            sB += bb[row][m0 + m + 10] - bb[row][m0 + m - 1];
            hA[row][m0 + m] = sA; hB[row][m0 + m] = sB;
        }
    }
    __syncthreads();

    // ---- stage 2 vertical + finalize: f = x - (box(a)*x + box(b)) ----
    // 32 cols x 8 segments of 4 rows = 256 threads. out row o uses hA o..o+10.
    {
        int col = tid >> 3, seg = tid & 7;
        int o0 = seg * 4;
        float sA = 0.0f, sB = 0.0f;
#pragma unroll
        for (int d = 0; d < KDIA; ++d) { sA += hA[o0 + d][col]; sB += hB[o0 + d][col]; }
#pragma unroll
        for (int k = 0; k < 4; ++k) {
            if (k) {
                sA += hA[o0 + k + 10][col] - hA[o0 + k - 1][col];
                sB += hB[o0 + k + 10][col] - hB[o0 + k - 1][col];
            }
            float x = xs[o0 + k + HALO][col + XOFF];
            fout[k] = x - (sA * INV_AREA * x + sB * INV_AREA);
        }
    }
}

template <bool USE_WS>
__global__ __launch_bounds__(NTHREADS) void guided_loss_kernel(
    const float* __restrict__ gen, const float* __restrict__ ir,
    const float* __restrict__ vi, float* __restrict__ partials,
    float* __restrict__ out)
{
    __shared__ float xs[REG][XSTR];
    __shared__ float hx[REG][MIDS];
    __shared__ float hxx[REG][MIDS];
    __shared__ float aa[MID][MIDS];
    __shared__ float bb[MID][MIDS];
    __shared__ float hA[MID][TS];
    __shared__ float hB[MID][TS];
    __shared__ float red[NTHREADS];

    const int tid = threadIdx.x;
    const int tileIdx = blockIdx.x;          // 0..255
    const int plane = blockIdx.y;            // 0..23
    const int tx = tileIdx & (TILES_X - 1);
    const int ty = tileIdx >> 4;
    const int py0 = ty * TILE, px0 = tx * TILE;
    const bool interior = (tx >= 1) && (tx <= TILES_X - 2) &&
                          (ty >= 1) && (ty <= TILES_Y - 2);
    const size_t poff = (size_t)plane * IMG_H * IMG_W;

    float f[4], flit[4];

    filter_tile(ir + poff, py0, px0, tid, interior, xs, hx, hxx, aa, bb, hA, hB, f);
#pragma unroll
    for (int k = 0; k < 4; ++k) flit[k] = fabsf(f[k]);

    filter_tile(vi + poff, py0, px0, tid, interior, xs, hx, hxx, aa, bb, hA, hB, f);
#pragma unroll
    for (int k = 0; k < 4; ++k) flit[k] = fmaxf(flit[k], fabsf(f[k]));

    filter_tile(gen + poff, py0, px0, tid, interior, xs, hx, hxx, aa, bb, hA, hB, f);
    float acc = 0.0f;
#pragma unroll
    for (int k = 0; k < 4; ++k) acc += fabsf(f[k] - flit[k]);

    // block reduction
    red[tid] = acc;
    __syncthreads();
    for (int off = NTHREADS / 2; off > 0; off >>= 1) {
        if (tid < off) red[tid] += red[tid + off];
        __syncthreads();
    }
    if (tid == 0) {
        if (USE_WS)
            partials[(size_t)blockIdx.y * gridDim.x + blockIdx.x] = red[0];
        else
            atomicAdd(out, red[0] * INV_N);
    }
}

__global__ __launch_bounds__(NTHREADS) void reduce_partials_kernel(
    const float* __restrict__ partials, int n, float* __restrict__ out)
{
    __shared__ float red[NTHREADS];
    float s = 0.0f;
    for (int i = threadIdx.x; i < n; i += NTHREADS) s += partials[i];
    red[threadIdx.x] = s;
    __syncthreads();
    for (int off = NTHREADS / 2; off > 0; off >>= 1) {
        if (threadIdx.x < off) red[threadIdx.x] += red[threadIdx.x + off];
        __syncthreads();
    }
    if (threadIdx.x == 0) out[0] = red[0] * INV_N;
}

extern "C" void kernel_launch(void* const* d_in, const int* in_sizes, int n_in,
                              void* d_out, int out_size, void* d_ws, size_t ws_size,
                              hipStream_t stream)
{
    (void)in_sizes; (void)n_in; (void)out_size;
    // setup_inputs order: general_img, ir_img, vi_img (all fp32)
    const float* gen = (const float*)d_in[0];
    const float* ir  = (const float*)d_in[1];
    const float* vi  = (const float*)d_in[2];
    float* out = (float*)d_out;

    dim3 grid(TILES_X * TILES_Y, NPLANES);
    const int nPartials = TILES_X * TILES_Y * NPLANES;  // 6144
    const size_t needWs = (size_t)nPartials * sizeof(float);

    if (ws_size >= needWs) {
        // deterministic two-stage reduction via workspace partials
        float* partials = (float*)d_ws;
        guided_loss_kernel<true><<<grid, NTHREADS, 0, stream>>>(gen, ir, vi, partials, out);
        reduce_partials_kernel<<<1, NTHREADS, 0, stream>>>(partials, nPartials, out);
    } else {
        (void)hipMemsetAsync(out, 0, sizeof(float), stream);
        guided_loss_kernel<false><<<grid, NTHREADS, 0, stream>>>(gen, ir, vi, nullptr, out);
    }
}